// JetGNN_2765958938745
// MI455X (gfx1250) — compile-verified
//
#include <hip/hip_runtime.h>

typedef __attribute__((ext_vector_type(2))) float v2f;
typedef __attribute__((ext_vector_type(8))) float v8f;

// D = A(16x4 f32) * B(4x16 f32) + C(16x16 f32), full fp32 precision.
__device__ __forceinline__ v8f wmma_f32_4(v2f a, v2f b, v8f c) {
  return __builtin_amdgcn_wmma_f32_16x16x4_f32(false, a, false, b, (short)0, c,
                                               false, false);
}

__global__ __launch_bounds__(256) void zero_f32(float* __restrict__ p, int n) {
  int i = blockIdx.x * 256 + threadIdx.x;
  if (i < n) p[i] = 0.0f;
}

// One lane per (edge, channel): coalesced gather of x[src] and coalesced
// atomic scatter into msg[dst]. Lane c==0 also accumulates the degree.
__global__ __launch_bounds__(256) void edge_scatter_c32(
    const int* __restrict__ ei, const float* __restrict__ x,
    float* __restrict__ msg, float* __restrict__ deg, int E) {
  int tid = blockIdx.x * 256 + threadIdx.x;
  if (tid >= E * 32) return;
  int e = tid >> 5, c = tid & 31;
  int s = ei[e];
  int d = ei[E + e];
  atomicAdd(&msg[(size_t)d * 32 + c], x[(size_t)s * 32 + c]);
  if (c == 0) atomicAdd(&deg[d], 1.0f);
}

__global__ __launch_bounds__(256) void edge_scatter_c64(
    const int* __restrict__ ei, const float* __restrict__ h,
    float* __restrict__ msg, int E) {
  int tid = blockIdx.x * 256 + threadIdx.x;
  if (tid >= E * 64) return;
  int e = tid >> 6, c = tid & 63;
  int s = ei[e];
  int d = ei[E + e];
  atomicAdd(&msg[(size_t)d * 64 + c], h[(size_t)s * 64 + c]);
}

__global__ __launch_bounds__(256) void count_nodes(const int* __restrict__ batch,
                                                   float* __restrict__ cnt, int N) {
  int i = blockIdx.x * 256 + threadIdx.x;
  if (i < N) atomicAdd(&cnt[batch[i]], 1.0f);
}

// One wave32 handles a 16-node tile: out16x64 = relu( (msg/deg)@Wl^T + b + feat@Wr^T ).
// K is the input feature width (32 for layer1, 64 for layer2).
// If POOL: instead of storing h, atomically accumulate into pool[batch[node]].
template <int K, bool POOL>
__global__ __launch_bounds__(256) void sage_layer(
    const float* __restrict__ feat,   // [N,K]
    const float* __restrict__ msg,    // [N,K] neighbor sums
    const float* __restrict__ deg,    // [N]
    const float* __restrict__ Wl,     // [64,K] row-major
    const float* __restrict__ Wr,     // [64,K] row-major
    const float* __restrict__ bias,   // [64]
    const int* __restrict__ batch,    // [N] (POOL only)
    float* __restrict__ out,          // [N,64] h  or  [G,64] pool
    int nTiles) {
  int wave = (int)((blockIdx.x * blockDim.x + threadIdx.x) >> 5);
  if (wave >= nTiles) return;  // wave-uniform: EXEC stays all-ones for WMMA
  int lane = threadIdx.x & 31;
  int half = lane >> 4;   // 0: K=0,1 ; 1: K=2,3 within each 4-wide K step
  int l16 = lane & 15;
  int rowA = wave * 16 + l16;

  constexpr int NK = K / 4;
  v2f am[NK], ax[NK];
  float invd = 1.0f / fmaxf(deg[rowA], 1.0f);
  const float* mrow = msg + (size_t)rowA * K + half * 2;
  const float* frow = feat + (size_t)rowA * K + half * 2;
#pragma unroll
  for (int k = 0; k < NK; ++k) {
    v2f m = *(const v2f*)(mrow + 4 * k);
    am[k].x = m.x * invd;
    am[k].y = m.y * invd;
    ax[k] = *(const v2f*)(frow + 4 * k);
  }

#pragma unroll
  for (int nt = 0; nt < 4; ++nt) {
    int n = nt * 16 + l16;  // output column handled by this lane
    float bv = bias[n];
    v8f c = {bv, bv, bv, bv, bv, bv, bv, bv};
    const float* wl = Wl + (size_t)n * K + half * 2;  // B = W^T fragments
    const float* wr = Wr + (size_t)n * K + half * 2;
#pragma unroll
    for (int k = 0; k < NK; ++k) {
      v2f b = *(const v2f*)(wl + 4 * k);
      c = wmma_f32_4(am[k], b, c);
    }
#pragma unroll
    for (int k = 0; k < NK; ++k) {
      v2f b = *(const v2f*)(wr + 4 * k);
      c = wmma_f32_4(ax[k], b, c);
    }
#pragma unroll
    for (int v = 0; v < 8; ++v) {
      float r = fmaxf(c[v], 0.0f);          // ReLU
      int node = wave * 16 + v + half * 8;  // C layout: VGPR v -> M=v / v+8
      if (POOL) {
        int g = batch[node];
        atomicAdd(&out[(size_t)g * 64 + n], r);
      } else {
        out[(size_t)node * 64 + n] = r;
      }
    }
  }
}

__global__ __launch_bounds__(128) void pooled_linear(
    const float* __restrict__ pool, const float* __restrict__ cnt,
    const float* __restrict__ W, const float* __restrict__ b,
    float* __restrict__ out, int G) {
  int tid = blockIdx.x * 128 + threadIdx.x;
  if (tid >= G * 2) return;
  int g = tid >> 1, o = tid & 1;
  float ic = 1.0f / fmaxf(cnt[g], 1.0f);
  float acc = b[o];
  const float* pr = pool + (size_t)g * 64;
  const float* wr = W + o * 64;
#pragma unroll
  for (int n = 0; n < 64; ++n) acc += pr[n] * ic * wr[n];
  out[tid] = acc;
}

extern "C" void kernel_launch(void* const* d_in, const int* in_sizes, int n_in,
                              void* d_out, int out_size, void* d_ws, size_t ws_size,
                              hipStream_t stream) {
  const float* x = (const float*)d_in[0];
  const int* ei = (const int*)d_in[1];
  const int* batch = (const int*)d_in[2];
  const float* W1l = (const float*)d_in[3];
  const float* b1 = (const float*)d_in[4];
  const float* W1r = (const float*)d_in[5];
  const float* W2l = (const float*)d_in[6];
  const float* b2 = (const float*)d_in[7];
  const float* W2r = (const float*)d_in[8];
  const float* Wlin = (const float*)d_in[9];
  const float* blin = (const float*)d_in[10];
  float* out = (float*)d_out;

  const int N = in_sizes[0] / 32;      // 200000
  const int E = in_sizes[1] / 2;       // 3200000
  const int G = out_size / 2;          // 4000
  const int nTiles = N / 16;           // 12500 (N divisible by 16)

  float* ws = (float*)d_ws;
  float* deg = ws;                         // [N]
  float* msg = deg + N;                    // [N,64] (used as [N,32] in layer 1)
  float* h1 = msg + (size_t)N * 64;        // [N,64]
  float* pool = h1 + (size_t)N * 64;       // [G,64]
  float* cnt = pool + (size_t)G * 64;      // [G]

  // --- zero accumulators (deg + msg32, pool + cnt) ---
  {
    int n1 = N + N * 32;
    zero_f32<<<(n1 + 255) / 256, 256, 0, stream>>>(ws, n1);
    int n2 = G * 64 + G;
    zero_f32<<<(n2 + 255) / 256, 256, 0, stream>>>(pool, n2);
  }

  // --- layer 1: scatter-mean + dual GEMM (WMMA f32) + ReLU -> h1 ---
  edge_scatter_c32<<<(E * 32 + 255) / 256, 256, 0, stream>>>(ei, x, msg, deg, E);
  sage_layer<32, false><<<(nTiles * 32 + 255) / 256, 256, 0, stream>>>(
      x, msg, deg, W1l, W1r, b1, nullptr, h1, nTiles);

  // --- layer 2: re-zero msg as [N,64], scatter h1, GEMM + ReLU + fused pool ---
  {
    int n3 = N * 64;
    zero_f32<<<(n3 + 255) / 256, 256, 0, stream>>>(msg, n3);
  }
  edge_scatter_c64<<<(E * 64 + 255) / 256, 256, 0, stream>>>(ei, h1, msg, E);
  count_nodes<<<(N + 255) / 256, 256, 0, stream>>>(batch, cnt, N);
  sage_layer<64, true><<<(nTiles * 32 + 255) / 256, 256, 0, stream>>>(
      h1, msg, deg, W2l, W2r, b2, batch, pool, nTiles);

  // --- global mean pool normalization + final linear 64->2 ---
  pooled_linear<<<(G * 2 + 127) / 128, 128, 0, stream>>>(pool, cnt, Wlin, blin,
                                                         out, G);
}